// PrivGCN_89807766159534
// MI455X (gfx1250) — compile-verified
//
#include <hip/hip_runtime.h>

// ---------------------------------------------------------------------------
// CDNA5 (gfx1250) GCN forward:  2x (GEMM -> normalized scatter-add -> ReLU)
//                               -> global mean pool -> linear head
// GEMMs use V_WMMA_F32_16X16X4_F32 (full fp32 matrix core, wave32), with the
// weight matrix repacked into WMMA-fragment order and staged into LDS via the
// Tensor Data Mover (tensor_load_to_lds + s_wait_tensorcnt).
// ---------------------------------------------------------------------------

typedef float v2f  __attribute__((ext_vector_type(2)));
typedef float v8f  __attribute__((ext_vector_type(8)));
typedef unsigned int u32x4 __attribute__((ext_vector_type(4)));
typedef int   i32x4 __attribute__((ext_vector_type(4)));
typedef int   i32x8 __attribute__((ext_vector_type(8)));

#define HID    128                 // feature width of both GEMMs (IN == HID)
#define NFRAG  (32 * 8)           // (K/4) k-steps x 8 column tiles
#define WP_F2  (NFRAG * 32)       // 8192 float2 = 64 KB repacked weights

// ---- fill a float buffer with a constant --------------------------------
__global__ void gcn_fill_f32(float* __restrict__ p, int n, float v) {
    int i = blockIdx.x * blockDim.x + threadIdx.x;
    if (i < n) p[i] = v;
}

// ---- count degrees (self-loop folded in via init to 1.0) ----------------
__global__ void gcn_deg_count(const int* __restrict__ dst, float* __restrict__ deg, int E) {
    int e = blockIdx.x * blockDim.x + threadIdx.x;
    if (e < E) atomicAdd(&deg[dst[e]], 1.0f);
}

// ---- deg -> deg^{-1/2} in place -----------------------------------------
__global__ void gcn_rsqrt(float* __restrict__ d, int n) {
    int i = blockIdx.x * blockDim.x + threadIdx.x;
    if (i < n) d[i] = rsqrtf(d[i]);
}

// ---- repack W[128x128] row-major -> fragment-major Wp[frag][lane] -------
// frag = (k0/4)*8 + j ; lane owns B-matrix element pair:
//   K = k0 + 2*(lane>>4) + {0,1},  N = j*16 + (lane&15)
// so each 16x4 B fragment is one contiguous, bank-conflict-free 256B block.
__global__ void gcn_repack_w(const float* __restrict__ W, float2* __restrict__ Wp) {
    int t = blockIdx.x * blockDim.x + threadIdx.x;   // 0 .. WP_F2-1
    if (t >= WP_F2) return;
    int lane = t & 31;
    int j    = (t >> 5) & 7;
    int kq   = t >> 8;                               // k0 / 4
    int half = lane >> 4, lrow = lane & 15;
    int k    = kq * 4 + 2 * half;
    int col  = j * 16 + lrow;
    float2 v;
    v.x = W[(size_t)k * HID + col];
    v.y = W[(size_t)(k + 1) * HID + col];
    Wp[t] = v;
}

// ---- H = X @ W via V_WMMA_F32_16X16X4_F32 -------------------------------
// 8 waves / block; one wave per 16-row strip, all 128 output columns.
// Repacked weights (64 KB) staged into LDS once per block by the TDM.
//
// Fragment layouts (ISA 7.12.2, 32-bit data, wave32):
//   A (16x4):  lane L<16:  M=L,    v0:K=k0+0, v1:K=k0+1
//              lane L>=16: M=L-16, v0:K=k0+2, v1:K=k0+3
//   B (4x16):  VGPR r: lanes 0-15 -> (K=k0+r, N=lane); lanes16-31 -> K=k0+r+2
//   C/D(16x16): VGPR r: lanes 0-15 -> M=r, lanes 16-31 -> M=r+8, N=lane%16
__global__ void gcn_gemm_wmma(const float2* __restrict__ Wp,
                              const float*  __restrict__ X,
                              float* __restrict__ H,
                              int nstrips) {
    __shared__ float2 WpLds[WP_F2];                 // 64 KB

    // ---- stage Wp into LDS with the Tensor Data Mover (wave 0 only) ----
    if (threadIdx.x < 32) {                         // wave-uniform branch
        unsigned lds_base = (unsigned)(unsigned long long)
            ((__attribute__((address_space(3))) const void*)WpLds);
        unsigned long long ga = (unsigned long long)Wp;
        // D# group0: count=1, lds_addr, global_addr[56:0], type=2
        u32x4 g0 = { 1u,
                     lds_base,
                     (unsigned)(ga & 0xffffffffull),
                     (unsigned)((ga >> 32) & 0x01ffffffull) | 0x80000000u };
        // D# group1: data_size=3 (8B); tensor_dim0 = tile_dim0 = 8192 elems,
        // tensor_dim1 = tile_dim1 = 1, dim0_stride = 8192.
        i32x8 g1 = { 0x30000,                        // data_size=3 @ [17:16]
                     (int)(8192u << 16),             // tensor_dim0 lo16 @ [31:16]
                     (int)(1u << 16),                // dim0 hi16=0 | tensor_dim1=1
                     (int)(8192u << 16),             // dim1 hi16=0 | tile_dim0=8192
                     1,                              // tile_dim1=1 | tile_dim2=0
                     8192,                           // dim0_stride lo32
                     0, 0 };                         // stride hi / dim1_stride
        i32x4 z4 = { 0, 0, 0, 0 };                   // groups 2/3 unused (<=2D)
        i32x8 z8 = { 0, 0, 0, 0, 0, 0, 0, 0 };
        __builtin_amdgcn_tensor_load_to_lds(g0, g1, z4, z4, z8, 0);
        __builtin_amdgcn_s_wait_tensorcnt(0);
    }
    __syncthreads();

    int wave = (int)blockIdx.x * (blockDim.x >> 5) + (threadIdx.x >> 5);
    if (wave >= nstrips) return;                    // wave-uniform: EXEC all-1s
    int lane = threadIdx.x & 31;
    int half = lane >> 4;
    int lrow = lane & 15;
    int row  = wave * 16 + lrow;
    const float* xrow = X + (size_t)row * HID;

    v8f c[8] = {};                                  // 8 col tiles, 64 VGPRs
    for (int k0 = 0; k0 < HID; k0 += 4) {
        v2f a;
        a.x = xrow[k0 + 2 * half + 0];
        a.y = xrow[k0 + 2 * half + 1];
        const float2* frag = &WpLds[((k0 >> 2) * 8) * 32 + lane];
        // hoist all 8 fragments first: lets the compiler clause the
        // ds_load_2addr_b64's and interleave s_wait_dscnt with WMMAs
        float2 t[8];
        #pragma unroll
        for (int j = 0; j < 8; ++j) t[j] = frag[j * 32];
        #pragma unroll
        for (int j = 0; j < 8; ++j) {
            v2f b; b.x = t[j].x; b.y = t[j].y;
            c[j] = __builtin_amdgcn_wmma_f32_16x16x4_f32(
                       /*neg_a=*/false, a, /*neg_b=*/false, b,
                       /*c_mod=*/(short)0, c[j],
                       /*reuse_a=*/false, /*reuse_b=*/false);
        }
    }
    // write back D: element (M = r + 8*half, N = j*16 + lrow)
    #pragma unroll
    for (int j = 0; j < 8; ++j) {
        #pragma unroll
        for (int r = 0; r < 8; ++r) {
            int m = r + 8 * half;
            H[(size_t)(wave * 16 + m) * HID + j * 16 + lrow] = c[j][r];
        }
    }
}

// ---- out[i,:] = h[i,:]*dinv[i]^2 + bias  (self-loop term + bias) --------
__global__ void gcn_init_out(const float* __restrict__ h,
                             const float* __restrict__ dinv,
                             const float* __restrict__ bias,
                             float* __restrict__ out, int n) {
    int i = blockIdx.x * blockDim.x + threadIdx.x;
    if (i >= n) return;
    int row = i >> 7;
    int col = i & (HID - 1);
    float di = dinv[row];
    out[i] = h[i] * di * di + bias[col];
}

// ---- edge scatter: out[d,:] += h[s,:] * dinv[s]*dinv[d] -----------------
__global__ void gcn_edge_scatter(const float* __restrict__ h,
                                 float* __restrict__ out,
                                 const int* __restrict__ src,
                                 const int* __restrict__ dst,
                                 const float* __restrict__ dinv, int E) {
    int gtid = blockIdx.x * blockDim.x + threadIdx.x;
    int e    = gtid >> 5;
    if (e >= E) return;
    int lane = threadIdx.x & 31;
    int s = src[e], d = dst[e];
    float w = dinv[s] * dinv[d];
    const float4* hs = (const float4*)(h + (size_t)s * HID);
    float4 v = hs[lane];
    float* o = out + (size_t)d * HID + lane * 4;
    atomicAdd(o + 0, v.x * w);
    atomicAdd(o + 1, v.y * w);
    atomicAdd(o + 2, v.z * w);
    atomicAdd(o + 3, v.w * w);
}

// ---- in-place ReLU ------------------------------------------------------
__global__ void gcn_relu(float* __restrict__ p, int n) {
    int i = blockIdx.x * blockDim.x + threadIdx.x;
    if (i < n) p[i] = fmaxf(p[i], 0.0f);
}

// ---- pool accumulation: pool[batch[row], col] += h[row, col] ------------
__global__ void gcn_pool_add(const float* __restrict__ h,
                             const int* __restrict__ batch,
                             float* __restrict__ pool, int n) {
    int i = blockIdx.x * blockDim.x + threadIdx.x;
    if (i >= n) return;
    int row = i >> 7;
    int col = i & (HID - 1);
    atomicAdd(&pool[(size_t)batch[row] * HID + col], h[i]);
}

__global__ void gcn_cnt_add(const int* __restrict__ batch,
                            float* __restrict__ cnt, int n) {
    int i = blockIdx.x * blockDim.x + threadIdx.x;
    if (i < n) atomicAdd(&cnt[batch[i]], 1.0f);
}

// ---- head: out[g,o] = (pool[g,:]/max(cnt[g],1)) @ Wl[:,o] + bl[o] -------
__global__ void gcn_head(const float* __restrict__ pool,
                         const float* __restrict__ cnt,
                         const float* __restrict__ Wl,
                         const float* __restrict__ bl,
                         float* __restrict__ out, int OUT) {
    int g = blockIdx.x;
    int o = threadIdx.x;
    float inv = 1.0f / fmaxf(cnt[g], 1.0f);
    float acc = bl[o];
    for (int h = 0; h < HID; ++h)
        acc = fmaf(pool[(size_t)g * HID + h] * inv, Wl[(size_t)h * OUT + o], acc);
    out[(size_t)g * OUT + o] = acc;
}

// ---------------------------------------------------------------------------
extern "C" void kernel_launch(void* const* d_in, const int* in_sizes, int n_in,
                              void* d_out, int out_size, void* d_ws, size_t ws_size,
                              hipStream_t stream) {
    const float* x     = (const float*)d_in[0];
    const int*   ei    = (const int*)  d_in[1];
    const int*   batch = (const int*)  d_in[2];
    const float* W1    = (const float*)d_in[3];
    const float* b1    = (const float*)d_in[4];
    const float* W2    = (const float*)d_in[5];
    const float* b2    = (const float*)d_in[6];
    const float* Wl    = (const float*)d_in[7];
    const float* bl    = (const float*)d_in[8];
    float*       out   = (float*)d_out;

    const int N   = in_sizes[2];          // 50000 nodes (batch vector length)
    const int E   = in_sizes[1] / 2;      // 800000 edges
    const int OUT = in_sizes[8];          // 64 (bl length)
    const int G   = out_size / OUT;       // 64 graphs

    const int* src = ei;
    const int* dst = ei + E;

    // workspace layout (floats): bufA | bufB | dinv | pool | cnt | Wp
    float*  ws   = (float*)d_ws;
    size_t  nh   = (size_t)N * HID;
    float*  bufA = ws;                     // GEMM outputs (h)
    float*  bufB = ws + nh;                // aggregation outputs
    float*  dinv = bufB + nh;              // deg -> deg^{-1/2}
    float*  pool = dinv + N;               // [G, HID]
    float*  cnt  = pool + (size_t)G * HID; // [G]
    float2* Wp   = (float2*)(cnt + G);     // repacked weights, 64 KB

    const int elems   = N * HID;           // 6.4M, fits int
    const int nstrips = N / 16;            // 3125 (N is a multiple of 16)
    const int gblk    = (nstrips + 7) / 8; // 8 waves (strips) per block
    dim3 blk(256);

    // --- symmetric normalization: dinv = rsqrt(1 + indegree) -------------
    gcn_fill_f32 <<<(N + 255) / 256, blk, 0, stream>>>(dinv, N, 1.0f);
    gcn_deg_count<<<(E + 255) / 256, blk, 0, stream>>>(dst, dinv, E);
    gcn_rsqrt    <<<(N + 255) / 256, blk, 0, stream>>>(dinv, N);

    // --- layer 1 ---------------------------------------------------------
    gcn_repack_w    <<<(WP_F2 + 255) / 256, blk, 0, stream>>>(W1, Wp);
    gcn_gemm_wmma   <<<gblk, 256, 0, stream>>>(Wp, x, bufA, nstrips);
    gcn_init_out    <<<(elems + 255) / 256, blk, 0, stream>>>(bufA, dinv, b1, bufB, elems);
    gcn_edge_scatter<<<(E * 32 + 255) / 256, blk, 0, stream>>>(bufA, bufB, src, dst, dinv, E);
    gcn_relu        <<<(elems + 255) / 256, blk, 0, stream>>>(bufB, elems);

    // --- layer 2 ---------------------------------------------------------
    gcn_repack_w    <<<(WP_F2 + 255) / 256, blk, 0, stream>>>(W2, Wp);
    gcn_gemm_wmma   <<<gblk, 256, 0, stream>>>(Wp, bufB, bufA, nstrips);
    gcn_init_out    <<<(elems + 255) / 256, blk, 0, stream>>>(bufA, dinv, b2, bufB, elems);
    gcn_edge_scatter<<<(E * 32 + 255) / 256, blk, 0, stream>>>(bufA, bufB, src, dst, dinv, E);
    gcn_relu        <<<(elems + 255) / 256, blk, 0, stream>>>(bufB, elems);

    // --- mean pool + linear head ----------------------------------------
    gcn_fill_f32<<<(G * HID + G + 255) / 256, blk, 0, stream>>>(pool, G * HID + G, 0.0f);
    gcn_pool_add<<<(elems + 255) / 256, blk, 0, stream>>>(bufB, batch, pool, elems);
    gcn_cnt_add <<<(N + 255) / 256, blk, 0, stream>>>(batch, cnt, N);
    gcn_head    <<<G, OUT, 0, stream>>>(pool, cnt, Wl, bl, out, OUT);
}